// FileRanker_61804579390128
// MI455X (gfx1250) — compile-verified
//
#include <hip/hip_runtime.h>
#include <hip/hip_bf16.h>

#define N_NODES 100000
#define N_EDGES 3200000
#define IN_CH   128
#define HID     64
#define BN_EPS  1e-5f

typedef __attribute__((ext_vector_type(2))) float v2f;
typedef __attribute__((ext_vector_type(8))) float v8f;

// ---------------------------------------------------------------------------
// Dual GEMM: P = H @ Wl^T, R = H @ Wr^T  (H:[n,CIN], W:[HID,CIN] row-major)
// One wave per 16-row tile; 8 waves/block -> 128 rows per block.
// Weights staged in LDS; V_WMMA_F32_16X16X4_F32 accumulation over K.
// ---------------------------------------------------------------------------
template<int CIN>
__global__ __launch_bounds__(256)
void gemm_dual_kernel(const float* __restrict__ H, const float* __restrict__ Wl,
                      const float* __restrict__ Wr, float* __restrict__ P,
                      float* __restrict__ Rout, int n)
{
    __shared__ float sW[2 * CIN * HID];
    for (int i = threadIdx.x; i < CIN * HID; i += 256) {
        sW[i]             = Wl[i];
        sW[CIN * HID + i] = Wr[i];
    }
    __syncthreads();

    const int wave    = threadIdx.x >> 5;
    const int lane    = threadIdx.x & 31;
    const int rowBase = (blockIdx.x * 8 + wave) * 16;
    if (rowBase >= n) return;                 // wave-uniform; EXEC stays all-ones

    const int m     = lane & 15;              // A: row in tile, B/D: col in tile
    const int khalf = (lane >> 4) * 2;        // K sub-offset for A/B fragments
    const float* sWl = sW;
    const float* sWr = sW + CIN * HID;

    v8f accL[4] = {};   // 4 column tiles of 16 -> 64 outputs (Wl)
    v8f accR[4] = {};   // same for Wr

    const float* hrow = H + (size_t)(rowBase + m) * CIN;
    #pragma unroll 4
    for (int k0 = 0; k0 < CIN; k0 += 4) {
        v2f a = *(const v2f*)(hrow + k0 + khalf);
        #pragma unroll
        for (int t = 0; t < 4; ++t) {
            v2f bl = *(const v2f*)(sWl + (t * 16 + m) * CIN + k0 + khalf);
            accL[t] = __builtin_amdgcn_wmma_f32_16x16x4_f32(
                false, a, false, bl, (short)0, accL[t], false, false);
            v2f br = *(const v2f*)(sWr + (t * 16 + m) * CIN + k0 + khalf);
            accR[t] = __builtin_amdgcn_wmma_f32_16x16x4_f32(
                false, a, false, br, (short)0, accR[t], false, false);
        }
    }

    const int rhalf = (lane >> 4) * 8;
    #pragma unroll
    for (int t = 0; t < 4; ++t) {
        #pragma unroll
        for (int r = 0; r < 8; ++r) {
            size_t o = (size_t)(rowBase + rhalf + r) * HID + t * 16 + m;
            P[o]    = accL[t][r];
            Rout[o] = accR[t][r];
        }
    }
}

// ---------------------------------------------------------------------------
// Head GEMM: O = relu(H @ W^T + bias)  (H:[n,64], W:[32,64])
// ---------------------------------------------------------------------------
__global__ __launch_bounds__(256)
void head1_kernel(const float* __restrict__ H, const float* __restrict__ W,
                  const float* __restrict__ bias, float* __restrict__ O, int n)
{
    __shared__ float sW[32 * 64];
    for (int i = threadIdx.x; i < 32 * 64; i += 256) sW[i] = W[i];
    __syncthreads();

    const int wave    = threadIdx.x >> 5;
    const int lane    = threadIdx.x & 31;
    const int rowBase = (blockIdx.x * 8 + wave) * 16;
    if (rowBase >= n) return;

    const int m     = lane & 15;
    const int khalf = (lane >> 4) * 2;
    v8f acc[2] = {};

    const float* hrow = H + (size_t)(rowBase + m) * 64;
    #pragma unroll 4
    for (int k0 = 0; k0 < 64; k0 += 4) {
        v2f a = *(const v2f*)(hrow + k0 + khalf);
        #pragma unroll
        for (int t = 0; t < 2; ++t) {
            v2f b = *(const v2f*)(sW + (t * 16 + m) * 64 + k0 + khalf);
            acc[t] = __builtin_amdgcn_wmma_f32_16x16x4_f32(
                false, a, false, b, (short)0, acc[t], false, false);
        }
    }

    const int rhalf = (lane >> 4) * 8;
    #pragma unroll
    for (int t = 0; t < 2; ++t) {
        float bv = bias[t * 16 + m];
        #pragma unroll
        for (int r = 0; r < 8; ++r) {
            float v = acc[t][r] + bv;
            O[(size_t)(rowBase + rhalf + r) * 32 + t * 16 + m] = fmaxf(v, 0.0f);
        }
    }
}

// ---------------------------------------------------------------------------
// Degree count (once): deg[dst[e]] += 1
// ---------------------------------------------------------------------------
__global__ __launch_bounds__(256)
void deg_kernel(const int* __restrict__ dst, float* __restrict__ deg, int nE)
{
    int i  = blockIdx.x * blockDim.x + threadIdx.x;
    int st = gridDim.x * blockDim.x;
    for (int e = i; e < nE; e += st)
        unsafeAtomicAdd(deg + dst[e], 1.0f);
}

// ---------------------------------------------------------------------------
// Edge scatter-add: S[dst[e], :] += P[src[e], :]  (one wave per edge,
// lane handles 2 consecutive features -> coalesced 256B gather,
// hardware global_atomic_add_f32 without return)
// ---------------------------------------------------------------------------
__global__ __launch_bounds__(256)
void scatter_kernel(const float* __restrict__ P, const int* __restrict__ src,
                    const int* __restrict__ dst, float* __restrict__ S, int nE)
{
    int gw   = (blockIdx.x * 256 + threadIdx.x) >> 5;
    int lane = threadIdx.x & 31;
    int nw   = (gridDim.x * 256) >> 5;
    for (int e = gw; e < nE; e += nw) {
        int s = src[e], d = dst[e];
        v2f v = *(const v2f*)(P + (size_t)s * HID + lane * 2);
        float* q = S + (size_t)d * HID + lane * 2;
        unsafeAtomicAdd(q,     v.x);
        unsafeAtomicAdd(q + 1, v.y);
    }
}

// ---------------------------------------------------------------------------
// Fused combine: Hout = relu(bn(S/max(deg,1) + bl + R))
// ---------------------------------------------------------------------------
__global__ __launch_bounds__(256)
void combine_kernel(const float* __restrict__ S, const float* __restrict__ Rm,
                    const float* __restrict__ deg, const float* __restrict__ bl,
                    const float* __restrict__ g, const float* __restrict__ b,
                    const float* __restrict__ mu, const float* __restrict__ var,
                    float* __restrict__ Hout, int n)
{
    int idx = blockIdx.x * 256 + threadIdx.x;
    if (idx >= n * HID) return;
    int node = idx >> 6, f = idx & 63;
    float dg = fmaxf(deg[node], 1.0f);
    float v  = S[idx] / dg + bl[f] + Rm[idx];
    v = (v - mu[f]) * rsqrtf(var[f] + BN_EPS) * g[f] + b[f];
    Hout[idx] = fmaxf(v, 0.0f);
}

// ---------------------------------------------------------------------------
// Final projection: out[i] = dot(HH[i, 0:32], hW2) + hb2
// ---------------------------------------------------------------------------
__global__ __launch_bounds__(256)
void head2_kernel(const float* __restrict__ HH, const float* __restrict__ hW2,
                  const float* __restrict__ hb2, float* __restrict__ out, int n)
{
    int i = blockIdx.x * 256 + threadIdx.x;
    if (i >= n) return;
    const float4* hp = (const float4*)(HH + (size_t)i * 32);
    const float4* wp = (const float4*)hW2;
    float acc = 0.0f;
    #pragma unroll
    for (int j = 0; j < 8; ++j) {
        float4 h4 = hp[j], w4 = wp[j];
        acc += h4.x * w4.x + h4.y * w4.y + h4.z * w4.z + h4.w * w4.w;
    }
    out[i] = acc + hb2[0];
}

// ---------------------------------------------------------------------------
extern "C" void kernel_launch(void* const* d_in, const int* in_sizes, int n_in,
                              void* d_out, int out_size, void* d_ws, size_t ws_size,
                              hipStream_t stream)
{
    const float* x   = (const float*)d_in[0];
    const int*   ei  = (const int*)  d_in[1];
    const float* Wl0 = (const float*)d_in[2];
    const float* Wr0 = (const float*)d_in[3];
    const float* bl0 = (const float*)d_in[4];
    const float* Wl1 = (const float*)d_in[5];
    const float* Wr1 = (const float*)d_in[6];
    const float* bl1 = (const float*)d_in[7];
    const float* Wl2 = (const float*)d_in[8];
    const float* Wr2 = (const float*)d_in[9];
    const float* bl2 = (const float*)d_in[10];
    const float* bng = (const float*)d_in[11];
    const float* bnb = (const float*)d_in[12];
    const float* bnm = (const float*)d_in[13];
    const float* bnv = (const float*)d_in[14];
    const float* hW1 = (const float*)d_in[15];
    const float* hb1 = (const float*)d_in[16];
    const float* hW2 = (const float*)d_in[17];
    const float* hb2 = (const float*)d_in[18];

    const int N = N_NODES, E = N_EDGES;
    const int* src = ei;          // edge_index[0, :]
    const int* dst = ei + E;      // edge_index[1, :]

    // workspace layout
    float* P   = (float*)d_ws;
    float* R   = P   + (size_t)N * HID;
    float* S   = R   + (size_t)N * HID;
    float* HA  = S   + (size_t)N * HID;
    float* HB  = HA  + (size_t)N * HID;
    float* deg = HB  + (size_t)N * HID;
    float* HH  = deg + N;         // [N, 32] head hidden

    const int gemmGrid = (N + 127) / 128;
    const int combGrid = (N * HID + 255) / 256;
    const int scatGrid = 4096;

    // degrees (edge-invariant: compute once per launch)
    hipMemsetAsync(deg, 0, (size_t)N * sizeof(float), stream);
    deg_kernel<<<2048, 256, 0, stream>>>(dst, deg, E);

    // ---- layer 0 (project-then-aggregate: 128ch -> 64ch before scatter) ----
    gemm_dual_kernel<IN_CH><<<gemmGrid, 256, 0, stream>>>(x, Wl0, Wr0, P, R, N);
    hipMemsetAsync(S, 0, (size_t)N * HID * sizeof(float), stream);
    scatter_kernel<<<scatGrid, 256, 0, stream>>>(P, src, dst, S, E);
    combine_kernel<<<combGrid, 256, 0, stream>>>(S, R, deg, bl0,
        bng + 0 * HID, bnb + 0 * HID, bnm + 0 * HID, bnv + 0 * HID, HA, N);

    // ---- layer 1 ----
    gemm_dual_kernel<HID><<<gemmGrid, 256, 0, stream>>>(HA, Wl1, Wr1, P, R, N);
    hipMemsetAsync(S, 0, (size_t)N * HID * sizeof(float), stream);
    scatter_kernel<<<scatGrid, 256, 0, stream>>>(P, src, dst, S, E);
    combine_kernel<<<combGrid, 256, 0, stream>>>(S, R, deg, bl1,
        bng + 1 * HID, bnb + 1 * HID, bnm + 1 * HID, bnv + 1 * HID, HB, N);

    // ---- layer 2 ----
    gemm_dual_kernel<HID><<<gemmGrid, 256, 0, stream>>>(HB, Wl2, Wr2, P, R, N);
    hipMemsetAsync(S, 0, (size_t)N * HID * sizeof(float), stream);
    scatter_kernel<<<scatGrid, 256, 0, stream>>>(P, src, dst, S, E);
    combine_kernel<<<combGrid, 256, 0, stream>>>(S, R, deg, bl2,
        bng + 2 * HID, bnb + 2 * HID, bnm + 2 * HID, bnv + 2 * HID, HA, N);

    // ---- MLP head ----
    head1_kernel<<<gemmGrid, 256, 0, stream>>>(HA, hW1, hb1, HH, N);
    head2_kernel<<<(N + 255) / 256, 256, 0, stream>>>(HH, hW2, hb2, (float*)d_out, N);
}